// SemiLoss_7834020348691
// MI455X (gfx1250) — compile-verified
//
#include <hip/hip_runtime.h>
#include <hip/hip_bf16.h>

// ---------------------------------------------------------------------------
// SemiLoss on MI455X (gfx1250, wave32).
// Strategy: convert column gather into a weighted full-row stream (w[c] =
// multiplicity of c in class_idx), one row per wave32 (lane l -> cols 4l..4l+3,
// b128 coalesced loads), skip masked-off rows (uniform per wave), and reduce
// each wave's partial with two V_WMMA_F32_16X16X4_F32 ops (matrix-pipe wave
// reduction). Deterministic two-stage block-partial reduction, no atomics.
// ---------------------------------------------------------------------------

typedef __attribute__((ext_vector_type(2))) float v2f;
typedef __attribute__((ext_vector_type(8))) float v8f;

#define NBLK      1024
#define THREADS   256
#define C_DIM     128
#define WS_W      0            // ws[0..127]          : column weights
#define WS_PART   C_DIM        // ws[128..128+NBLK)   : per-block partials
#define WS_MASK   (C_DIM+NBLK) // ws[128+NBLK]        : float(sum(batch_mask))

// Full wave32 sum via the WMMA pipe; returns the sum broadcast to all lanes.
// WMMA1: A(m,0)=p[m], A(m,2)=p[m+16] (A VGPR0=p, VGPR1=0), B=ones
//   -> D(m,n) = p[m]+p[m+16] = s[m]
//   D layout: lane n<16 holds s[j] in component j; lane n>=16 holds s[8+j].
//   t = sum_j d[j]  => lanes 0-15: sum(s[0..7]); lanes 16-31: sum(s[8..15]).
// WMMA2: A(m,0)=t[m], A(m,2)=t[m+16], B=ones -> D2 = T_lo + T_hi = total.
__device__ __forceinline__ float wave_sum_wmma(float p) {
    v2f a;  a[0] = p;    a[1] = 0.0f;
    v2f b1; b1[0] = 1.0f; b1[1] = 1.0f;
    v8f c = {};
    v8f d = __builtin_amdgcn_wmma_f32_16x16x4_f32(
        false, a, false, b1, (short)0, c, false, false);
    float t = ((d[0] + d[1]) + (d[2] + d[3])) + ((d[4] + d[5]) + (d[6] + d[7]));
    v2f a2; a2[0] = t; a2[1] = 0.0f;
    v8f d2 = __builtin_amdgcn_wmma_f32_16x16x4_f32(
        false, a2, false, b1, (short)0, c, false, false);
    return d2[0];
}

__device__ __forceinline__ float loss_elem(float o, float oc, float y) {
    // sharp = o>0.5 ? o + (1-o)/4 : o - o/4
    float sharp = (o > 0.5f) ? fmaf(1.0f - o, 0.25f, o) : o * 0.75f;
    float lo  = fmaxf(__logf(o),        -100.0f);  // clamped log
    float l1m = fmaxf(__logf(1.0f - o), -100.0f);
    float bce = -(y * lo + (1.0f - y) * l1m);
    float dp = o - sharp;
    float dc = oc - sharp;
    return bce + dp * dp + dc * dc;
}

// ---- Kernel 1: column weights w[c] from class_idx (int64), mask popcount ---
__global__ void semi_prep(const long long* __restrict__ cls,
                          const unsigned char* __restrict__ mask,
                          float* __restrict__ ws, int B, int K) {
    __shared__ int cnt[C_DIM];
    __shared__ int msum[THREADS];
    const int t = threadIdx.x;
    if (t < C_DIM) cnt[t] = 0;
    __syncthreads();
    if (t < K) atomicAdd(&cnt[(int)cls[t]], 1);
    __syncthreads();
    if (t < C_DIM) ws[WS_W + t] = (float)cnt[t];

    // batch_mask is a byte array of 0/1; count via 4-byte popcount stream.
    const unsigned int* m4 = (const unsigned int*)mask;
    const int n4 = B >> 2;
    int s = 0;
    for (int i = t; i < n4; i += THREADS)
        s += __popc(m4[i] & 0x01010101u);
    msum[t] = s;
    __syncthreads();
    for (int off = THREADS / 2; off > 0; off >>= 1) {
        if (t < off) msum[t] += msum[t + off];
        __syncthreads();
    }
    if (t == 0) ws[WS_MASK] = (float)msum[0];
}

// ---- Kernel 2: main streaming loss, one row per wave32 ---------------------
__global__ void __launch_bounds__(THREADS)
semi_main(const float* __restrict__ o,
          const float* __restrict__ oc,
          const float* __restrict__ y,
          const unsigned char* __restrict__ mask,
          const float* __restrict__ ws,
          float* __restrict__ partials, int B) {
    const int lane   = threadIdx.x & 31;
    const int wib    = threadIdx.x >> 5;                       // wave in block
    const int gwid   = (blockIdx.x * THREADS + threadIdx.x) >> 5;
    const int nwaves = (gridDim.x * THREADS) >> 5;
    const int col    = lane << 2;                              // 4 cols / lane

    // Loop-invariant per-lane column weights (the whole "gather").
    const float4 w4 = *(const float4*)(ws + WS_W + col);

    float acc = 0.0f;
    for (int b = gwid; b < B; b += nwaves) {
        if (mask[b]) {                       // uniform across the wave
            const size_t base = (size_t)b * C_DIM + col;
            const float4 o4 = *(const float4*)(o  + base);
            const float4 c4 = *(const float4*)(oc + base);
            const float4 y4 = *(const float4*)(y  + base);
            acc += w4.x * loss_elem(o4.x, c4.x, y4.x);
            acc += w4.y * loss_elem(o4.y, c4.y, y4.y);
            acc += w4.z * loss_elem(o4.z, c4.z, y4.z);
            acc += w4.w * loss_elem(o4.w, c4.w, y4.w);
        }
    }

    // Matrix-pipe wave reduction (EXEC all ones here: no divergence).
    const float wsum = wave_sum_wmma(acc);

    __shared__ float lds[THREADS / 32];
    if (lane == 0) lds[wib] = wsum;
    __syncthreads();
    if (threadIdx.x == 0) {
        float s = 0.0f;
        #pragma unroll
        for (int i = 0; i < THREADS / 32; ++i) s += lds[i];
        partials[blockIdx.x] = s;            // deterministic, no atomics
    }
}

// ---- Kernel 3: fixed-order final reduction + divide ------------------------
__global__ void semi_final(const float* __restrict__ partials,
                           const float* __restrict__ ws,
                           float* __restrict__ out, int nblk, float Kf) {
    __shared__ float s[THREADS];
    const int t = threadIdx.x;
    float a = 0.0f;
    for (int i = t; i < nblk; i += THREADS) a += partials[i];
    s[t] = a;
    __syncthreads();
    for (int off = THREADS / 2; off > 0; off >>= 1) {
        if (t < off) s[t] += s[t + off];
        __syncthreads();
    }
    if (t == 0) out[0] = s[0] / (ws[WS_MASK] * Kf);
}

extern "C" void kernel_launch(void* const* d_in, const int* in_sizes, int n_in,
                              void* d_out, int out_size, void* d_ws, size_t ws_size,
                              hipStream_t stream) {
    const float*         o    = (const float*)d_in[0];          // (B,C) f32
    const float*         oc   = (const float*)d_in[1];          // (B,C) f32
    const float*         y    = (const float*)d_in[2];          // (B,C) f32
    const long long*     cls  = (const long long*)d_in[3];      // (K,) int64
    const unsigned char* mask = (const unsigned char*)d_in[4];  // (B,) bool

    const int K = in_sizes[3];
    const int B = in_sizes[4];

    float* ws  = (float*)d_ws;
    float* out = (float*)d_out;

    semi_prep <<<1,    THREADS, 0, stream>>>(cls, mask, ws, B, K);
    semi_main <<<NBLK, THREADS, 0, stream>>>(o, oc, y, mask, ws, ws + WS_PART, B);
    semi_final<<<1,    THREADS, 0, stream>>>(ws + WS_PART, ws, out, NBLK, (float)K);
}